// GNN_34823594836467
// MI455X (gfx1250) — compile-verified
//
#include <hip/hip_runtime.h>

typedef __attribute__((ext_vector_type(2))) float v2f;
typedef __attribute__((ext_vector_type(8))) float v8f;

#define DIM 128
#define LDS_STRIDE 130   // pad 128 -> 130 floats: row-strided WMMA fragment reads hit distinct LDS banks
#define TILE_M 128

// h = dropout(relu(x @ W^T + b)) using V_WMMA_F32_16X16X4_F32
__global__ __launch_bounds__(256) void gnn_linear_wmma(
    const float* __restrict__ x, const float* __restrict__ W,
    const float* __restrict__ bias, const float* __restrict__ mask_u,
    float* __restrict__ h, int n_nodes)
{
    __shared__ float Xl[TILE_M * LDS_STRIDE];
    __shared__ float Wl[DIM * LDS_STRIDE];

    const int tid = threadIdx.x;
    const int m0  = blockIdx.x * TILE_M;

    // Stage W [128 x 128] into LDS (coalesced, padded rows)
    for (int i = tid; i < DIM * DIM; i += 256) {
        int r = i >> 7, c = i & 127;
        Wl[r * LDS_STRIDE + c] = W[i];
    }
    // Stage X tile [128 x 128] into LDS, zero-padding out-of-range rows
    for (int i = tid; i < TILE_M * DIM; i += 256) {
        int r = i >> 7, c = i & 127;
        int gm = m0 + r;
        Xl[r * LDS_STRIDE + c] = (gm < n_nodes) ? x[(long long)gm * DIM + c] : 0.0f;
    }
    __syncthreads();

    const int wave = tid >> 5;       // 8 waves: wave w owns M-rows [w*16, w*16+16)
    const int lane = tid & 31;
    const int half = lane >> 4;      // 0: lanes 0-15, 1: lanes 16-31
    const int mrow = lane & 15;

    // A fragment (16x4 f32, 2 VGPRs): lane supplies row (lane%16), k = k0 + 2*half + {0,1}
    const float* arow = &Xl[(wave * 16 + mrow) * LDS_STRIDE + 2 * half];

    for (int nt = 0; nt < 8; ++nt) {
        // B[k][n] = W[n][k]  ->  same float2-per-lane indexing as A, reading W's row n
        const float* brow = &Wl[(nt * 16 + mrow) * LDS_STRIDE + 2 * half];

        v8f acc = {};
        #pragma unroll
        for (int k0 = 0; k0 < DIM; k0 += 4) {
            v2f a = *(const v2f*)(arow + k0);
            v2f b = *(const v2f*)(brow + k0);
            // (neg_a, A, neg_b, B, c_mod, C, reuse_a, reuse_b)
            acc = __builtin_amdgcn_wmma_f32_16x16x4_f32(
                false, a, false, b, (short)0, acc, false, false);
        }

        // C/D layout: VGPR r -> M = r + 8*half, N = lane%16 (within tile)
        const int ncol = nt * 16 + mrow;
        const float bv = bias[ncol];
        #pragma unroll
        for (int r = 0; r < 8; ++r) {
            int gm = m0 + wave * 16 + r + 8 * half;
            if (gm < n_nodes) {
                float v = acc[r] + bv;
                v = v > 0.0f ? v : 0.0f;                                  // relu
                v = (mask_u[(long long)gm * DIM + ncol] >= 0.5f) ? v * 2.0f : 0.0f; // dropout p=0.5
                h[(long long)gm * DIM + ncol] = v;
            }
        }
    }
}

__global__ void gnn_zero(float* __restrict__ out, int n)
{
    int i = blockIdx.x * blockDim.x + threadIdx.x;
    if (i < n) out[i] = 0.0f;
}

// out[dst[e]] += h[src[e]] ; one thread per (edge, feature). h and out are
// L2-resident (25.6 MB each in 192 MB L2) -> atomics resolve at L2.
__global__ void gnn_scatter_add(const float* __restrict__ h,
                                const int* __restrict__ src,
                                const int* __restrict__ dst,
                                float* __restrict__ out, int n_edges)
{
    long long idx = (long long)blockIdx.x * blockDim.x + threadIdx.x;
    int e = (int)(idx >> 7);
    int t = (int)(idx & 127);
    if (e < n_edges) {
        int s = src[e];
        int d = dst[e];
        float v = h[(long long)s * DIM + t];
        atomicAdd(&out[(long long)d * DIM + t], v);
    }
}

extern "C" void kernel_launch(void* const* d_in, const int* in_sizes, int n_in,
                              void* d_out, int out_size, void* d_ws, size_t ws_size,
                              hipStream_t stream)
{
    const float* x      = (const float*)d_in[0];
    const float* W      = (const float*)d_in[1];
    const float* b      = (const float*)d_in[2];
    const float* mask_u = (const float*)d_in[3];
    const int*   src    = (const int*)d_in[4];
    const int*   dst    = (const int*)d_in[5];
    float*       out    = (float*)d_out;
    float*       h      = (float*)d_ws;   // 50000*128*4 = 25.6 MB scratch

    const int n_nodes = in_sizes[0] / DIM;
    const int n_edges = in_sizes[4];

    // 1) fused linear+relu+dropout via WMMA
    int gemm_blocks = (n_nodes + TILE_M - 1) / TILE_M;
    gnn_linear_wmma<<<gemm_blocks, 256, 0, stream>>>(x, W, b, mask_u, h, n_nodes);

    // 2) zero the output accumulator
    gnn_zero<<<(out_size + 255) / 256, 256, 0, stream>>>(out, out_size);

    // 3) edge scatter-add (L2-resident atomics)
    long long total = (long long)n_edges * DIM;
    int sblocks = (int)((total + 255) / 256);
    gnn_scatter_add<<<sblocks, 256, 0, stream>>>(h, src, dst, out, n_edges);
}